// VectorQuantizer_32040456028670
// MI455X (gfx1250) — compile-verified
//
#include <hip/hip_runtime.h>
#include <hip/hip_bf16.h>

typedef __attribute__((ext_vector_type(2))) float v2f;
typedef __attribute__((ext_vector_type(8))) float v8f;
typedef int vec4i __attribute__((vector_size(16)));

#define NVEC   131072      // B*H*W
#define DIM    64
#define KCB    1024
#define NBLK   2048        // NVEC / 64
#define LDSP   68          // padded LDS row stride (floats): conflict-free b64 reads

#define _AS1 __attribute__((address_space(1)))
#define _AS3 __attribute__((address_space(3)))

// ---- CDNA5 async global->LDS (ASYNCcnt path), builtin if present else asm ----
__device__ __forceinline__ void async_b128(const float* g, float* l) {
#if __has_builtin(__builtin_amdgcn_global_load_async_to_lds_b128)
  __builtin_amdgcn_global_load_async_to_lds_b128(
      (_AS1 vec4i*)(uintptr_t)g,
      (_AS3 vec4i*)(uint32_t)(uintptr_t)l, 0, 0);
#else
  asm volatile("global_load_async_to_lds_b128 %0, %1, off"
               :: "v"((uint32_t)(uintptr_t)l),
                  "v"((unsigned long long)(uintptr_t)g)
               : "memory");
#endif
}

__device__ __forceinline__ void async_b32(const float* g, float* l) {
#if __has_builtin(__builtin_amdgcn_global_load_async_to_lds_b32)
  __builtin_amdgcn_global_load_async_to_lds_b32(
      (_AS1 int*)(uintptr_t)g,
      (_AS3 int*)(uint32_t)(uintptr_t)l, 0, 0);
#else
  asm volatile("global_load_async_to_lds_b32 %0, %1, off"
               :: "v"((uint32_t)(uintptr_t)l),
                  "v"((unsigned long long)(uintptr_t)g)
               : "memory");
#endif
}

__device__ __forceinline__ void wait_async0() {
#if __has_builtin(__builtin_amdgcn_s_wait_asynccnt)
  __builtin_amdgcn_s_wait_asynccnt(0);
#else
  asm volatile("s_wait_asynccnt 0x0" ::: "memory");
#endif
}

// ---------------------------------------------------------------------------
// Kernel 1: codebook row squared norms
// ---------------------------------------------------------------------------
__global__ __launch_bounds__(128) void vq_cnorm_kernel(
    const float* __restrict__ cb, float* __restrict__ cnorm) {
  int k = blockIdx.x * 128 + threadIdx.x;   // 0..1023
  float s = 0.f;
  const float* row = cb + (size_t)k * DIM;
#pragma unroll 8
  for (int d = 0; d < DIM; ++d) { float v = row[d]; s += v * v; }
  cnorm[k] = s;
}

// ---------------------------------------------------------------------------
// Kernel 2: main VQ kernel. One block = one (b,h) pair = 64 vectors of dim 64.
// 128 threads = 4 waves; each wave owns a 16-row slab and runs the WMMA search
// with 4 independent accumulator chains (sub-loop force-unrolled).
// Codebook tiles double-buffered via async-to-LDS; x transposed via async b32.
// ---------------------------------------------------------------------------
__global__ __launch_bounds__(128) void vq_main_kernel(
    const float* __restrict__ x, const float* __restrict__ cb,
    const float* __restrict__ cnorm, float* __restrict__ out,
    float* __restrict__ partial) {
  __shared__ float xs[64][LDSP];       // x tile, transposed: xs[w][d]
  __shared__ float cs[2][64][LDSP];    // double-buffered codebook tile
  __shared__ float cns[2][64];         // double-buffered codebook norms
  __shared__ int   bidx_s[64];         // winning index per vector
  __shared__ float red[128];           // loss reduction

  const int t    = threadIdx.x;         // 0..127
  const int lane = t & 31;
  const int wave = t >> 5;              // 0..3
  const int m    = lane & 15;           // N-lane / M-lane within 16
  const int hi   = lane >> 4;           // half-wave select
  const int koff = hi * 2;              // f32 WMMA 16x16x4: lanes0-15 K={0,1}, lanes16-31 K={2,3}

  const int b = blockIdx.x >> 6;        // 0..31
  const int h = blockIdx.x & 63;        // 0..63
  const size_t xbase = (size_t)b * (DIM * 64 * 64) + (size_t)h * 64;  // + d*4096 + w

  // ---- async stage x tile (transpose [d][w] -> xs[w][d]); per-lane LDS dest ----
#pragma unroll
  for (int it = 0; it < 32; ++it) {
    int d = it * 2 + (t >> 6);
    int w = t & 63;
    async_b32(x + xbase + (size_t)d * 4096 + w, &xs[w][d]);
  }
  // ---- async stage codebook tile 0 into buffer 0 ----
#pragma unroll
  for (int it = 0; it < 8; ++it) {
    int kr = it * 8 + (t >> 4);
    int d  = (t & 15) * 4;
    async_b128(cb + (size_t)kr * DIM + d, &cs[0][kr][d]);
  }
  if (t < 64) cns[0][t] = cnorm[t];

  wait_async0();
  __syncthreads();

  // ---- preload A fragments for this wave's 16 rows (16 steps of K=4) ----
  const int arow = wave * 16 + m;
  v2f afrag[16];
#pragma unroll
  for (int s = 0; s < 16; ++s)
    afrag[s] = *(const v2f*)&xs[arow][4 * s + koff];

  float best[8];
  int   bidx[8];
#pragma unroll
  for (int j = 0; j < 8; ++j) { best[j] = 3.4e38f; bidx[j] = 0; }

  // ---- search all 1024 codebook rows, 64 at a time, double-buffered ----
  for (int kt = 0; kt < 16; ++kt) {
    const int buf = kt & 1;
    if (kt) {
      wait_async0();       // my async loads of tile kt are in LDS
      __syncthreads();     // everyone's loads done + prior compute on buf finished
    }
    if (kt < 15) {         // fire tile kt+1 into the other buffer
#pragma unroll
      for (int it = 0; it < 8; ++it) {
        int kr = it * 8 + (t >> 4);
        int d  = (t & 15) * 4;
        async_b128(cb + (size_t)((kt + 1) * 64 + kr) * DIM + d,
                   &cs[buf ^ 1][kr][d]);
      }
      if (t < 64) cns[buf ^ 1][t] = cnorm[(kt + 1) * 64 + t];
    }

#pragma unroll
    for (int sub = 0; sub < 4; ++sub) {   // force 4 independent WMMA chains
      const int crow = sub * 16 + m;          // B column n = lane&15
      const float cn = cns[buf][crow];
      v8f acc = {0.f, 0.f, 0.f, 0.f, 0.f, 0.f, 0.f, 0.f};
#pragma unroll
      for (int s = 0; s < 16; ++s) {
        v2f bfrag = *(const v2f*)&cs[buf][crow][4 * s + koff];
        acc = __builtin_amdgcn_wmma_f32_16x16x4_f32(
            false, afrag[s], false, bfrag, (short)0, acc, false, false);
      }
      const int kg = kt * 64 + sub * 16 + m;  // codebook index this lane scored
#pragma unroll
      for (int j = 0; j < 8; ++j) {
        float v = cn - 2.0f * acc[j];         // ||c||^2 - 2 x.c  (||x||^2 constant)
        if (v < best[j]) { best[j] = v; bidx[j] = kg; }
      }
    }
  }

  // ---- cross-lane argmin within each 16-lane N group (first-index ties) ----
#pragma unroll
  for (int j = 0; j < 8; ++j) {
    float bv = best[j];
    int   bi = bidx[j];
#pragma unroll
    for (int mask = 1; mask < 16; mask <<= 1) {
      float ov = __shfl_xor(bv, mask, 32);
      int   oi = __shfl_xor(bi, mask, 32);
      if (ov < bv || (ov == bv && oi < bi)) { bv = ov; bi = oi; }
    }
    if (m == 0) bidx_s[wave * 16 + j + 8 * hi] = bi;  // row j + 8*hi of this slab
  }
  __syncthreads();

  // ---- gather codebook winners, write NCHW output, accumulate loss ----
  float lsum = 0.f;
#pragma unroll 4
  for (int it = 0; it < 32; ++it) {
    int d = it * 2 + (t >> 6);
    int w = t & 63;
    float q  = cb[(size_t)bidx_s[w] * DIM + d];
    float xv = xs[w][d];
    float df = q - xv;
    lsum += df * df;
    out[xbase + (size_t)d * 4096 + w] = q;
  }

  red[t] = lsum;
  __syncthreads();
  for (int off = 64; off > 0; off >>= 1) {
    if (t < off) red[t] += red[t + off];
    __syncthreads();
  }
  if (t == 0) partial[blockIdx.x] = red[0];
}

// ---------------------------------------------------------------------------
// Kernel 3: deterministic final loss reduction
// ---------------------------------------------------------------------------
__global__ __launch_bounds__(256) void vq_loss_kernel(
    const float* __restrict__ partial, float* __restrict__ loss_out) {
  __shared__ float red[256];
  int t = threadIdx.x;
  float s = 0.f;
  for (int i = t; i < NBLK; i += 256) s += partial[i];
  red[t] = s;
  __syncthreads();
  for (int off = 128; off > 0; off >>= 1) {
    if (t < off) red[t] += red[t + off];
    __syncthreads();
  }
  if (t == 0)
    loss_out[0] = 1.25f * red[0] * (1.0f / (float)(NVEC * (double)DIM));
}

// ---------------------------------------------------------------------------
extern "C" void kernel_launch(void* const* d_in, const int* in_sizes, int n_in,
                              void* d_out, int out_size, void* d_ws, size_t ws_size,
                              hipStream_t stream) {
  const float* x  = (const float*)d_in[0];   // [32,64,64,64] fp32
  const float* cb = (const float*)d_in[1];   // [1024,64] fp32
  float* out = (float*)d_out;                // 8388608 quantized + 1 loss

  float* cnorm   = (float*)d_ws;             // 1024 floats
  float* partial = cnorm + KCB;              // 2048 floats

  vq_cnorm_kernel<<<KCB / 128, 128, 0, stream>>>(cb, cnorm);
  vq_main_kernel<<<NBLK, 128, 0, stream>>>(x, cb, cnorm, out, partial);
  vq_loss_kernel<<<1, 256, 0, stream>>>(partial, out + (size_t)NVEC * DIM);
}